// GazeKLD_27891517620669
// MI455X (gfx1250) — compile-verified
//
#include <hip/hip_runtime.h>

#define EPS 1e-6f
#define B_ROWS (1 << 20)       // 1048576 rows
#define BLOCKS 1024
#define THREADS 256

typedef float v2f __attribute__((ext_vector_type(2)));
typedef float v4f __attribute__((ext_vector_type(4)));
typedef float v8f __attribute__((ext_vector_type(8)));

// Wave32 full-precision sum of one float per lane using V_WMMA_F32_16X16X4_F32.
// A (16x4 f32, 2 VGPRs/lane): VGPR0 = lane value -> A[m,0] (lanes 0-15) / A[m,2]
// (lanes 16-31); VGPR1 = 0 -> A[m,1]/A[m,3]. B = ones(4x16) so
// D[m,n] = x_m + x_{m+16} for every column n. D's 8 VGPRs in lane l hold rows
// 0-7 (l<16) or 8-15 (l>=16); summing them + one xor-16 shuffle yields the
// full 32-lane sum in every lane. Requires EXEC == all ones (uniform flow).
__device__ __forceinline__ float wave_sum32(float x) {
    v2f a; a.x = x;    a.y = 0.0f;
    v2f b; b.x = 1.0f; b.y = 1.0f;
    v8f c = {0.0f, 0.0f, 0.0f, 0.0f, 0.0f, 0.0f, 0.0f, 0.0f};
    v8f d = __builtin_amdgcn_wmma_f32_16x16x4_f32(
        /*neg_a=*/false, a, /*neg_b=*/false, b,
        /*c_mod=*/(short)0, c, /*reuse_a=*/false, /*reuse_b=*/false);
    float h = ((d[0] + d[1]) + (d[2] + d[3])) + ((d[4] + d[5]) + (d[6] + d[7]));
    h += __shfl_xor(h, 16, 32);   // combine the two half-wave row groups
    return h;
}

__device__ __forceinline__ float block_sum256(float v, float* sm8) {
    float w = wave_sum32(v);
    const int warp = threadIdx.x >> 5;
    const int lane = threadIdx.x & 31;
    if (lane == 0) sm8[warp] = w;
    __syncthreads();
    float t = 0.0f;
    if (threadIdx.x == 0) {
#pragma unroll
        for (int i = 0; i < 8; ++i) t += sm8[i];
    }
    return t;   // valid in thread 0 only
}

// Kernel 1: stream `true` (128 B/row) + `pred` (16 B/row) with non-temporal
// B128 loads, compute per-row KL in f32, reduce to one partial per block.
__global__ __launch_bounds__(THREADS) void kld_rows_kernel(
    const float* __restrict__ pred, const float* __restrict__ truep,
    float* __restrict__ ws) {
    const int tid = blockIdx.x * THREADS + threadIdx.x;
    const int stride = BLOCKS * THREADS;           // 262144 -> 4 rows/thread
    float acc = 0.0f;

#pragma unroll
    for (int it = 0; it < B_ROWS / (BLOCKS * THREADS); ++it) {
        const int b = tid + it * stride;
        const v4f* tp = (const v4f*)truep + (size_t)b * 8;

        float s1x = 0.0f, s1y = 0.0f, s2x = 0.0f, s2y = 0.0f;
#pragma unroll
        for (int i = 0; i < 8; ++i) {
            v4f v = __builtin_nontemporal_load(tp + i);  // (t,c) pairs: x,z = ch0; y,w = ch1
            s1x += v.x + v.z;
            s1y += v.y + v.w;
            s2x += v.x * v.x + v.z * v.z;
            s2y += v.y * v.y + v.w * v.w;
        }
        v4f p = __builtin_nontemporal_load((const v4f*)pred + b);

        const float pv0 = p.z + EPS, pv1 = p.w + EPS;
        const float tm0 = s1x * 0.0625f, tm1 = s1y * 0.0625f;
        const float tv0 = (s2x - s1x * s1x * 0.0625f) * (1.0f / 15.0f) + EPS;
        const float tv1 = (s2y - s1y * s1y * 0.0625f) * (1.0f / 15.0f) + EPS;
        const float dm0 = p.x - tm0, dm1 = p.y - tm1;

        acc += 0.5f * (__logf(pv0 / tv0) + __logf(pv1 / tv1)
                       + tv0 / pv0 + tv1 / pv1
                       + dm0 * dm0 / pv0 + dm1 * dm1 / pv1 - 2.0f);
    }

    __shared__ float sm8[8];
    float t = block_sum256(acc, sm8);
    if (threadIdx.x == 0) ws[blockIdx.x] = t;
}

// Kernel 2: deterministic reduction of the 1024 block partials, scale by 1/B.
__global__ __launch_bounds__(THREADS) void kld_final_kernel(
    const float* __restrict__ ws, float* __restrict__ out) {
    float acc = 0.0f;
#pragma unroll
    for (int i = 0; i < BLOCKS / THREADS; ++i)
        acc += ws[threadIdx.x + i * THREADS];

    __shared__ float sm8[8];
    float t = block_sum256(acc, sm8);
    if (threadIdx.x == 0) out[0] = t * (1.0f / (float)B_ROWS);
}

extern "C" void kernel_launch(void* const* d_in, const int* in_sizes, int n_in,
                              void* d_out, int out_size, void* d_ws, size_t ws_size,
                              hipStream_t stream) {
    const float* pred  = (const float*)d_in[0];   // (B, 4)   f32
    const float* truep = (const float*)d_in[1];   // (B,16,2) f32
    float* ws = (float*)d_ws;                     // needs 1024 * 4 B = 4 KB

    kld_rows_kernel<<<BLOCKS, THREADS, 0, stream>>>(pred, truep, ws);
    kld_final_kernel<<<1, THREADS, 0, stream>>>(ws, (float*)d_out);
}